// _Attention_4844723109923
// MI455X (gfx1250) — compile-verified
//
#include <hip/hip_runtime.h>
#include <math.h>
#include <stdint.h>

#define DIMC 512
#define NHEADS 8
#define HDIM 64
#define BATCH 64
#define SEQ 256
#define BN (BATCH * SEQ)      // 16384 rows
#define QKVN (3 * DIMC)       // 1536 cols
#define RELBINS 961

typedef float v2f __attribute__((ext_vector_type(2)));
typedef float v8f __attribute__((ext_vector_type(8)));

// D(16x16,f32) = A(16x4,f32) * B(4x16,f32) + C   -- exact fp32 path on CDNA5
__device__ __forceinline__ v8f wmma4(v2f a, v2f b, v8f c) {
  return __builtin_amdgcn_wmma_f32_16x16x4_f32(
      /*neg_a=*/false, a, /*neg_b=*/false, b,
      /*c_mod=*/(short)0, c, /*reuse_a=*/false, /*reuse_b=*/false);
}

// Async copy 8 bytes global -> LDS (ASYNCcnt-tracked, GV addressing mode).
__device__ __forceinline__ void async_g2l_b64(unsigned lds_off,
                                              const float* gptr) {
  asm volatile("global_load_async_to_lds_b64 %0, %1, off"
               :
               : "v"(lds_off), "v"(gptr)
               : "memory");
}
__device__ __forceinline__ void wait_async0() {
  asm volatile("s_wait_asynccnt 0x0" ::: "memory");
}

// ---------------------------------------------------------------------------
// Kernel 1: expand rel_pos_bias[h, rel_pos_index[n,m]] -> biasx[h][n][m]
// ---------------------------------------------------------------------------
__global__ __launch_bounds__(256) void bias_expand_kernel(
    const int* __restrict__ rel_idx, const float* __restrict__ rel_bias,
    float* __restrict__ biasx) {
  int idx = blockIdx.x * blockDim.x + threadIdx.x;  // [0, 8*65536)
  int hh = idx >> 16;
  int nm = idx & 65535;
  biasx[idx] = rel_bias[hh * RELBINS + rel_idx[nm]];
}

// ---------------------------------------------------------------------------
// Kernel 2: qkv = x @ qkv_w^T + qkv_b, scattered to q/k/v [B,H,N,D].
// Scale 1/sqrt(D)=0.125 folded into q. 8 waves/WG, 32x32 tile per wave.
// ---------------------------------------------------------------------------
__global__ __launch_bounds__(256) void qkv_kernel(
    const float* __restrict__ x, const float* __restrict__ w,
    const float* __restrict__ bias, float* __restrict__ qb,
    float* __restrict__ kb, float* __restrict__ vb) {
  const int lane = threadIdx.x & 31;
  const int wave = threadIdx.x >> 5;
  const int l16 = lane & 15;
  const int hi = lane >> 4;
  const int klo = hi * 2;
  const int M0 = blockIdx.x * 64 + (wave >> 2) * 32;
  const int N0 = blockIdx.y * 128 + (wave & 3) * 32;

  v8f acc[2][2] = {};
  const float* xa0 = x + (size_t)(M0 + l16) * DIMC + klo;
  const float* xa1 = x + (size_t)(M0 + 16 + l16) * DIMC + klo;
  const float* wb0 = w + (size_t)(N0 + l16) * DIMC + klo;
  const float* wb1 = w + (size_t)(N0 + 16 + l16) * DIMC + klo;
#pragma unroll 4
  for (int k = 0; k < DIMC; k += 4) {
    v2f a0 = *(const v2f*)(xa0 + k);
    v2f a1 = *(const v2f*)(xa1 + k);
    v2f b0 = *(const v2f*)(wb0 + k);
    v2f b1 = *(const v2f*)(wb1 + k);
    acc[0][0] = wmma4(a0, b0, acc[0][0]);
    acc[0][1] = wmma4(a0, b1, acc[0][1]);
    acc[1][0] = wmma4(a1, b0, acc[1][0]);
    acc[1][1] = wmma4(a1, b1, acc[1][1]);
  }

#pragma unroll
  for (int u = 0; u < 2; ++u) {
    const int c = N0 + u * 16 + l16;
    const float bc = bias[c];
    const int which = c >> 9;        // 0=q 1=k 2=v
    const int cc = c & 511;
    const int hh = cc >> 6;
    const int dd = cc & 63;
    float* dst = (which == 0) ? qb : ((which == 1) ? kb : vb);
    const float sc = (which == 0) ? 0.125f : 1.0f;
#pragma unroll
    for (int t = 0; t < 2; ++t) {
#pragma unroll
      for (int r = 0; r < 8; ++r) {
        const int m = M0 + t * 16 + r + 8 * hi;
        const int bb = m >> 8;
        const int nn = m & 255;
        dst[(size_t)((bb * NHEADS + hh) * SEQ + nn) * HDIM + dd] =
            (acc[t][u][r] + bc) * sc;
      }
    }
  }
}

// ---------------------------------------------------------------------------
// Kernel 3: fused flash attention per (b,h). 8 waves, 32 rows each.
// K/V blocks staged to LDS via GLOBAL_LOAD_ASYNC_TO_LDS (ASYNCcnt).
// ---------------------------------------------------------------------------
__global__ __launch_bounds__(256) void attn_kernel(
    const float* __restrict__ qbuf, const float* __restrict__ kbuf,
    const float* __restrict__ vbuf, const float* __restrict__ biasx,
    float* __restrict__ ctx) {
  __shared__ __align__(16) float plds[8][2][16][18];  // per-wave P tiles
  __shared__ __align__(16) float Kl[16][66];          // K block (padded)
  __shared__ __align__(16) float Vl[16][66];          // V block (padded)

  const int tid = threadIdx.x;
  const int lane = tid & 31;
  const int wave = tid >> 5;
  const int l16 = lane & 15;
  const int hi = lane >> 4;
  const int klo = hi * 2;
  const int bh = blockIdx.x;  // b*8+h
  const int b = bh >> 3;
  const int h = bh & 7;
  const int m0 = wave * 32;

  const float* Q = qbuf + (size_t)bh * SEQ * HDIM;
  const float* K = kbuf + (size_t)bh * SEQ * HDIM;
  const float* V = vbuf + (size_t)bh * SEQ * HDIM;
  const float* Bb = biasx + (size_t)h * SEQ * SEQ;

  // each thread copies two 8B chunks per tile: rows 0..7 then 8..15
  const int r0 = tid >> 5;        // 0..7
  const int w0 = tid & 31;        // 8-byte chunk within 64-float row
  const int r1 = r0 + 8;
  const unsigned kd0 = (unsigned)(size_t)&Kl[r0][w0 * 2];
  const unsigned kd1 = (unsigned)(size_t)&Kl[r1][w0 * 2];
  const unsigned vd0 = (unsigned)(size_t)&Vl[r0][w0 * 2];
  const unsigned vd1 = (unsigned)(size_t)&Vl[r1][w0 * 2];

  v8f o[2][4] = {};
  v8f mrow[2], srow[2];
#pragma unroll
  for (int t = 0; t < 2; ++t)
#pragma unroll
    for (int r = 0; r < 8; ++r) {
      mrow[t][r] = -__builtin_inff();
      srow[t][r] = 0.0f;
    }

  for (int j = 0; j < SEQ / 16; ++j) {
    // ---- async-stage K/V block j into LDS ----
    const float* Kg = K + (size_t)j * 16 * HDIM;
    const float* Vg = V + (size_t)j * 16 * HDIM;
    async_g2l_b64(kd0, Kg + r0 * HDIM + w0 * 2);
    async_g2l_b64(kd1, Kg + r1 * HDIM + w0 * 2);
    async_g2l_b64(vd0, Vg + r0 * HDIM + w0 * 2);
    async_g2l_b64(vd1, Vg + r1 * HDIM + w0 * 2);
    wait_async0();
    __syncthreads();

    // ---- S = (Q*scale) @ K^T over K-dim 64 (K from LDS) ----
    v8f st[2] = {};
    const float* Q0 = Q + (size_t)(m0 + l16) * HDIM + klo;
    const float* Q1 = Q + (size_t)(m0 + 16 + l16) * HDIM + klo;
#pragma unroll
    for (int kk = 0; kk < HDIM; kk += 4) {
      v2f bq = *(const v2f*)&Kl[l16][kk + klo];
      v2f a0 = *(const v2f*)(Q0 + kk);
      v2f a1 = *(const v2f*)(Q1 + kk);
      st[0] = wmma4(a0, bq, st[0]);
      st[1] = wmma4(a1, bq, st[1]);
    }

#pragma unroll
    for (int t = 0; t < 2; ++t) {
      // bias add (element (row=m0+t*16+r+8*hi, col=j*16+l16))
#pragma unroll
      for (int r = 0; r < 8; ++r)
        st[t][r] += Bb[(size_t)(m0 + t * 16 + r + 8 * hi) * SEQ + j * 16 + l16];

      // block row-max across 16 columns (16 lanes of this half-wave)
      v8f bm = st[t];
#pragma unroll
      for (int mask = 1; mask <= 8; mask <<= 1)
#pragma unroll
        for (int r = 0; r < 8; ++r)
          bm[r] = fmaxf(bm[r], __shfl_xor(bm[r], mask, 32));

      v8f mnew, corr, p;
#pragma unroll
      for (int r = 0; r < 8; ++r) {
        mnew[r] = fmaxf(mrow[t][r], bm[r]);
        corr[r] = __expf(mrow[t][r] - mnew[r]);
        mrow[t][r] = mnew[r];
      }
#pragma unroll
      for (int r = 0; r < 8; ++r) p[r] = __expf(st[t][r] - mnew[r]);

      v8f ps = p;
#pragma unroll
      for (int mask = 1; mask <= 8; mask <<= 1)
#pragma unroll
        for (int r = 0; r < 8; ++r) ps[r] += __shfl_xor(ps[r], mask, 32);

#pragma unroll
      for (int r = 0; r < 8; ++r)
        srow[t][r] = srow[t][r] * corr[r] + ps[r];
#pragma unroll
      for (int nt = 0; nt < 4; ++nt)
#pragma unroll
        for (int r = 0; r < 8; ++r) o[t][nt][r] *= corr[r];

      // stage P (C-layout) to LDS so it can be reread in A-fragment layout
      float* pb = &plds[wave][t][0][0];
#pragma unroll
      for (int r = 0; r < 8; ++r) pb[(r + 8 * hi) * 18 + l16] = p[r];
    }

    // ---- O += P @ V_block (K=16 keys, 4 k-steps of 4; V from LDS) ----
#pragma unroll
    for (int kk = 0; kk < 16; kk += 4) {
      v2f a0 = *(const v2f*)&plds[wave][0][l16][kk + klo];
      v2f a1 = *(const v2f*)&plds[wave][1][l16][kk + klo];
#pragma unroll
      for (int nt = 0; nt < 4; ++nt) {
        v2f bv = {Vl[kk + klo][nt * 16 + l16], Vl[kk + klo + 1][nt * 16 + l16]};
        o[0][nt] = wmma4(a0, bv, o[0][nt]);
        o[1][nt] = wmma4(a1, bv, o[1][nt]);
      }
    }
    __syncthreads();  // protect K/V LDS tiles before next iteration's copies
  }

  // normalize + write ctx[b, n, h*64+d]
#pragma unroll
  for (int t = 0; t < 2; ++t)
#pragma unroll
    for (int nt = 0; nt < 4; ++nt)
#pragma unroll
      for (int r = 0; r < 8; ++r) {
        const int m = m0 + t * 16 + r + 8 * hi;
        const int d = nt * 16 + l16;
        ctx[(size_t)(b * SEQ + m) * DIMC + h * HDIM + d] =
            o[t][nt][r] / srow[t][r];
      }
}

// ---------------------------------------------------------------------------
// Kernel 4: out = ctx @ proj_w^T + proj_b
// ---------------------------------------------------------------------------
__global__ __launch_bounds__(256) void proj_kernel(
    const float* __restrict__ ctx, const float* __restrict__ w,
    const float* __restrict__ bias, float* __restrict__ out) {
  const int lane = threadIdx.x & 31;
  const int wave = threadIdx.x >> 5;
  const int l16 = lane & 15;
  const int hi = lane >> 4;
  const int klo = hi * 2;
  const int M0 = blockIdx.x * 64 + (wave >> 2) * 32;
  const int N0 = blockIdx.y * 128 + (wave & 3) * 32;

  v8f acc[2][2] = {};
  const float* xa0 = ctx + (size_t)(M0 + l16) * DIMC + klo;
  const float* xa1 = ctx + (size_t)(M0 + 16 + l16) * DIMC + klo;
  const float* wb0 = w + (size_t)(N0 + l16) * DIMC + klo;
  const float* wb1 = w + (size_t)(N0 + 16 + l16) * DIMC + klo;
#pragma unroll 4
  for (int k = 0; k < DIMC; k += 4) {
    v2f a0 = *(const v2f*)(xa0 + k);
    v2f a1 = *(const v2f*)(xa1 + k);
    v2f b0 = *(const v2f*)(wb0 + k);
    v2f b1 = *(const v2f*)(wb1 + k);
    acc[0][0] = wmma4(a0, b0, acc[0][0]);
    acc[0][1] = wmma4(a0, b1, acc[0][1]);
    acc[1][0] = wmma4(a1, b0, acc[1][0]);
    acc[1][1] = wmma4(a1, b1, acc[1][1]);
  }

#pragma unroll
  for (int u = 0; u < 2; ++u) {
    const int c = N0 + u * 16 + l16;
    const float bc = bias[c];
#pragma unroll
    for (int t = 0; t < 2; ++t)
#pragma unroll
      for (int r = 0; r < 8; ++r) {
        const int m = M0 + t * 16 + r + 8 * hi;
        out[(size_t)m * DIMC + c] = acc[t][u][r] + bc;
      }
  }
}

// ---------------------------------------------------------------------------
extern "C" void kernel_launch(void* const* d_in, const int* in_sizes, int n_in,
                              void* d_out, int out_size, void* d_ws,
                              size_t ws_size, hipStream_t stream) {
  (void)in_sizes; (void)n_in; (void)out_size; (void)ws_size;
  const float* x        = (const float*)d_in[0];
  const int*   relidx   = (const int*)d_in[1];
  const float* qkv_w    = (const float*)d_in[2];
  const float* qkv_b    = (const float*)d_in[3];
  const float* proj_w   = (const float*)d_in[4];
  const float* proj_b   = (const float*)d_in[5];
  const float* rel_bias = (const float*)d_in[6];
  float* out = (float*)d_out;

  // workspace layout (floats): q | k | v | ctx | biasx  (~136 MB)
  const size_t BHND = (size_t)BATCH * NHEADS * SEQ * HDIM;  // 8388608
  float* ws = (float*)d_ws;
  float* qb = ws;
  float* kb = qb + BHND;
  float* vb = kb + BHND;
  float* ctx = vb + BHND;
  float* biasx = ctx + (size_t)BN * DIMC;

  bias_expand_kernel<<<dim3((NHEADS * SEQ * SEQ) / 256), dim3(256), 0, stream>>>(
      relidx, rel_bias, biasx);
  qkv_kernel<<<dim3(BN / 64, QKVN / 128), dim3(256), 0, stream>>>(
      x, qkv_w, qkv_b, qb, kb, vb);
  attn_kernel<<<dim3(BATCH * NHEADS), dim3(256), 0, stream>>>(
      qb, kb, vb, biasx, ctx);
  proj_kernel<<<dim3(BN / 64, DIMC / 128), dim3(256), 0, stream>>>(
      ctx, proj_w, proj_b, out);
}